// NeuralODE_59837484368132
// MI455X (gfx1250) — compile-verified
//
#include <hip/hip_runtime.h>
#include <hip/hip_bf16.h>
#include <math.h>

typedef __attribute__((ext_vector_type(2))) float v2f;
typedef __attribute__((ext_vector_type(8))) float v8f;

#define NDIM   5
#define NSAVE  64
#define MAXSTEPS 64

// ---- LDS layout (float offsets) ----
#define OFF_TE 0        // 64  t_eval
#define OFF_B4 64       // 8   b4 (5 used)
#define OFF_B1 72       // 64
#define OFF_B2 136      // 64
#define OFF_B3 200      // 64
#define OFF_A1 264      // 4 mtiles * 2 chunks * 64 = 512   (mstride = 2)
#define OFF_A2 776      // 4 mtiles * 16 chunks * 64 = 4096 (mstride = 16)
#define OFF_A3 4872     // 4096
#define OFF_A4 8968     // 1 mtile * 16 chunks * 64 = 1024
#define OFF_ACT 9992    // per-wave activation buffers, 3200 floats each
#define ACT_PER_WAVE 3200
// per-wave: act0 (+0, 128), act1 (+128, 1024), act2 (+1152, 1024), act3 (+2176, 1024)
#define LDS_FLOATS (OFF_ACT + 4 * ACT_PER_WAVE)   // 22792 floats
#define LDS_BYTES  (LDS_FLOATS * 4)               // 91168 bytes

// ---- Tsit5 tableau ----
#define C_A21 0.161f
#define C_A31 -0.008480655492356989f
#define C_A32 0.335480655492357f
#define C_A41 2.8971530571054935f
#define C_A42 -6.359448489975075f
#define C_A43 4.3622954328695815f
#define C_A51 5.325864828439257f
#define C_A52 -11.748883564062828f
#define C_A53 7.4955393428898365f
#define C_A54 -0.09249506636175525f
#define C_A61 5.86145544294642f
#define C_A62 -12.92096931784711f
#define C_A63 8.159367898576159f
#define C_A64 -0.071584973281401f
#define C_A65 -0.028269050394068383f
#define C_B1 0.09646076681806523f
#define C_B2 0.01f
#define C_B3 0.4798896504144996f
#define C_B4 1.379008574103742f
#define C_B5 -3.290069515436081f
#define C_B6 2.324710524099774f
#define C_E1 -0.00178001105222577714f
#define C_E2 -0.0008164344596567469f
#define C_E3 0.007880878010261995f
#define C_E4 -0.1447110071732629f
#define C_E5 0.5823571654525552f
#define C_E6 -0.45808210592918697f
#define C_E7 0.015151515151515152f
#define C_RTOL 1e-3f
#define C_ATOL 1e-6f
#define C_SAFETY 0.9f
#define C_FMIN 0.2f
#define C_FMAX 10.0f
#define C_DT0 0.1f
#define C_EPS 1e-6f

#define LOG2E 1.44269504088896340736f
#define LN2   0.69314718055994530942f

__device__ __forceinline__ v8f wmma4(v2f a, v2f b, v8f c) {
    // D = A(16x4 f32) x B(4x16 f32) + C(16x16 f32)
    return __builtin_amdgcn_wmma_f32_16x16x4_f32(
        false, a, false, b, (short)0, c, false, false);
}

__device__ __forceinline__ float softplus_f(float x) {
    // logaddexp(x,0) = max(x,0) + ln2*log2(1 + exp2(-|x|*log2e))
    // log2 argument in [1,2] -> raw v_exp_f32/v_log_f32, no denorm fixups
    float e = __builtin_amdgcn_exp2f(-fabsf(x) * LOG2E);
    float l = __builtin_amdgcn_logf(1.0f + e) * LN2;
    return fmaxf(x, 0.0f) + l;
}

__device__ __forceinline__ float bcast16(float v, int lane) {
    // every lane reads from lane (lane & 15): replicate low half into high half
    int r = __builtin_amdgcn_ds_bpermute((lane & 15) << 2, __float_as_int(v));
    return __int_as_float(r);
}

// Generic dense layer (transposed): out = softplus(W @ in + b).
// Activations & weights live in LDS in chunk-slot packing:
//   slot index = chunk*64 + lane*2 + p, K-slot (2*half+p) of chunk (t,j)
//   corresponds to feature t*16 + half*8 + 2*j + p.
// Compile-time NCH/MSTRIDE -> scalar (uniform) loop control; noinline keeps
// a single out-of-line copy per instantiation.
template <int NCH, int MSTRIDE>
__device__ __noinline__ void dense_layer(int lane, int aOff, int bOff,
                                         int inOff, int outOff) {
    extern __shared__ float lds[];
    const int hlf = lane >> 4;
    const int l2  = lane * 2;

    v8f acc[4];
#pragma unroll
    for (int m = 0; m < 4; ++m)
#pragma unroll
        for (int r = 0; r < 8; ++r)
            acc[m][r] = lds[bOff + m * 16 + hlf * 8 + r];

#pragma unroll 4
    for (int ch = 0; ch < NCH; ++ch) {
        v2f B  = *(const v2f*)&lds[inOff + ch * 64 + l2];
        v2f A0 = *(const v2f*)&lds[aOff + (0 * MSTRIDE + ch) * 64 + l2];
        v2f A1 = *(const v2f*)&lds[aOff + (1 * MSTRIDE + ch) * 64 + l2];
        v2f A2 = *(const v2f*)&lds[aOff + (2 * MSTRIDE + ch) * 64 + l2];
        v2f A3 = *(const v2f*)&lds[aOff + (3 * MSTRIDE + ch) * 64 + l2];
        acc[0] = wmma4(A0, B, acc[0]);
        acc[1] = wmma4(A1, B, acc[1]);
        acc[2] = wmma4(A2, B, acc[2]);
        acc[3] = wmma4(A3, B, acc[3]);
    }

    // softplus + store in chunk-slot packing (becomes next layer's B operands)
#pragma unroll
    for (int m = 0; m < 4; ++m)
#pragma unroll
        for (int j = 0; j < 4; ++j) {
            v2f s;
            s[0] = softplus_f(acc[m][2 * j]);
            s[1] = softplus_f(acc[m][2 * j + 1]);
            *(v2f*)&lds[outOff + (m * 4 + j) * 64 + l2] = s;
        }
}

// Full MLP vector field f(y): 5 -> 64 -> 64 -> 64 -> 5.
__device__ __forceinline__ void mlp_stage(int lane, int wAct,
                                          const float yin[NDIM],
                                          float kout[NDIM]) {
    extern __shared__ float lds[];
    const int hlf = lane >> 4;
    const int l2  = lane * 2;

    // pack y (K=5 padded to 8 -> 2 chunks) into act0
    {
        v2f c0, c1;
        c0[0] = hlf ? yin[2] : yin[0];
        c0[1] = hlf ? yin[3] : yin[1];
        c1[0] = hlf ? 0.0f : yin[4];
        c1[1] = 0.0f;
        *(v2f*)&lds[wAct + 0 * 64 + l2] = c0;
        *(v2f*)&lds[wAct + 1 * 64 + l2] = c1;
    }
    dense_layer<2, 2>(lane, OFF_A1, OFF_B1, wAct, wAct + 128);
    dense_layer<16, 16>(lane, OFF_A2, OFF_B2, wAct + 128, wAct + 1152);
    dense_layer<16, 16>(lane, OFF_A3, OFF_B3, wAct + 1152, wAct + 2176);

    // ---- layer 4 (64 -> 5): 4 split accumulator chains, rolled loop ----
    v8f p[4];
#pragma unroll
    for (int q = 0; q < 4; ++q)
#pragma unroll
        for (int r = 0; r < 8; ++r) p[q][r] = 0.0f;
#pragma unroll 1
    for (int ch = 0; ch < 16; ch += 4) {
#pragma unroll
        for (int q = 0; q < 4; ++q) {
            v2f B = *(const v2f*)&lds[wAct + 2176 + (ch + q) * 64 + l2];
            v2f A = *(const v2f*)&lds[OFF_A4 + (ch + q) * 64 + l2];
            p[q] = wmma4(A, B, p[q]);
        }
    }
    v8f o = p[0] + p[1] + p[2] + p[3];
#pragma unroll
    for (int d = 0; d < NDIM; ++d)
        kout[d] = bcast16(o[d] + lds[OFF_B4 + d], lane);
}

__global__ __launch_bounds__(128, 4) void NeuralODE_tsit5_wmma_kernel(
    const float* __restrict__ y0g, const float* __restrict__ teg,
    const float* __restrict__ W1, const float* __restrict__ b1,
    const float* __restrict__ W2, const float* __restrict__ b2,
    const float* __restrict__ W3, const float* __restrict__ b3,
    const float* __restrict__ W4, const float* __restrict__ b4,
    float* __restrict__ out) {
    extern __shared__ float lds[];

    const int tid  = threadIdx.x;
    const int wave = tid >> 5;
    const int lane = tid & 31;
    const int l15  = lane & 15;

    // ---------------- stage weights/biases into LDS (A-operand packing) ----
    for (int i = tid; i < 64; i += 128) {
        lds[OFF_TE + i] = teg[i];
        lds[OFF_B1 + i] = b1[i];
        lds[OFF_B2 + i] = b2[i];
        lds[OFF_B3 + i] = b3[i];
    }
    if (tid < 8) lds[OFF_B4 + tid] = (tid < NDIM) ? b4[tid] : 0.0f;

    // layer 1 A-pack: [m(4)][c(2)][lane(32)][p(2)]
    for (int e = tid; e < 512; e += 128) {
        int m = e >> 7, c = (e >> 6) & 1, l = (e >> 1) & 31, pp = e & 1;
        int o = m * 16 + (l & 15), hf = l >> 4;
        float v;
        if (c == 0) v = W1[o * 5 + (2 * hf + pp)];
        else        v = (hf == 0 && pp == 0) ? W1[o * 5 + 4] : 0.0f;
        lds[OFF_A1 + e] = v;
    }
    // layers 2/3 A-pack: [m(4)][chunk(16)][lane(32)][p(2)]
    for (int e = tid; e < 4096; e += 128) {
        int m = e >> 10, ch = (e >> 6) & 15, l = (e >> 1) & 31, pp = e & 1;
        int o = m * 16 + (l & 15), hf = l >> 4;
        int t = ch >> 2, j = ch & 3;
        int fin = t * 16 + hf * 8 + 2 * j + pp;
        lds[OFF_A2 + e] = W2[o * 64 + fin];
        lds[OFF_A3 + e] = W3[o * 64 + fin];
    }
    // layer 4 A-pack: [chunk(16)][lane(32)][p(2)], rows >= 5 zero-padded
    for (int e = tid; e < 1024; e += 128) {
        int ch = e >> 6, l = (e >> 1) & 31, pp = e & 1;
        int o = l & 15, hf = l >> 4;
        int t = ch >> 2, j = ch & 3;
        int fin = t * 16 + hf * 8 + 2 * j + pp;
        lds[OFF_A4 + e] = (o < NDIM) ? W4[o * 64 + fin] : 0.0f;
    }
    __syncthreads();

    // ---------------- per-wave state: 16 batch elements -------------------
    const int tileBase = blockIdx.x * 64 + wave * 16;
    const int bidx = tileBase + l15;  // batch element of this lane (replicated)
    const int wAct = OFF_ACT + wave * ACT_PER_WAVE;

    float y[NDIM];
#pragma unroll
    for (int d = 0; d < NDIM; ++d) y[d] = y0g[bidx * NDIM + d];

    // zero-fill this wave's output region, then write s=0 <- y0
    {
        float* ob = out + (size_t)tileBase * (NSAVE * NDIM);
        for (int i = lane; i < 16 * NSAVE * NDIM; i += 32) ob[i] = 0.0f;
        if (lane < 16) {
#pragma unroll
            for (int d = 0; d < NDIM; ++d)
                out[((size_t)bidx * NSAVE + 0) * NDIM + d] = y[d];
        }
    }

    float tcur = lds[OFF_TE + 0];
    const float t1 = lds[OFF_TE + NSAVE - 1];
    float h = C_DT0;

    float k1[NDIM], k2[NDIM], k3[NDIM], k4[NDIM], k5[NDIM], k6[NDIM], k7[NDIM];
    float ytmp[NDIM], ynew[NDIM];

#pragma unroll 1
    for (int step = 0; step < MAXSTEPS; ++step) {
        const bool done = tcur >= t1 - C_EPS;
        const float hc = done ? 0.0f : fminf(h, t1 - tcur);

        mlp_stage(lane, wAct, y, k1);
#pragma unroll
        for (int d = 0; d < NDIM; ++d) ytmp[d] = y[d] + hc * (C_A21 * k1[d]);
        mlp_stage(lane, wAct, ytmp, k2);
#pragma unroll
        for (int d = 0; d < NDIM; ++d)
            ytmp[d] = y[d] + hc * (C_A31 * k1[d] + C_A32 * k2[d]);
        mlp_stage(lane, wAct, ytmp, k3);
#pragma unroll
        for (int d = 0; d < NDIM; ++d)
            ytmp[d] = y[d] + hc * (C_A41 * k1[d] + C_A42 * k2[d] + C_A43 * k3[d]);
        mlp_stage(lane, wAct, ytmp, k4);
#pragma unroll
        for (int d = 0; d < NDIM; ++d)
            ytmp[d] = y[d] + hc * (C_A51 * k1[d] + C_A52 * k2[d] +
                                   C_A53 * k3[d] + C_A54 * k4[d]);
        mlp_stage(lane, wAct, ytmp, k5);
#pragma unroll
        for (int d = 0; d < NDIM; ++d)
            ytmp[d] = y[d] + hc * (C_A61 * k1[d] + C_A62 * k2[d] + C_A63 * k3[d] +
                                   C_A64 * k4[d] + C_A65 * k5[d]);
        mlp_stage(lane, wAct, ytmp, k6);
#pragma unroll
        for (int d = 0; d < NDIM; ++d)
            ynew[d] = y[d] + hc * (C_B1 * k1[d] + C_B2 * k2[d] + C_B3 * k3[d] +
                                   C_B4 * k4[d] + C_B5 * k5[d] + C_B6 * k6[d]);
        mlp_stage(lane, wAct, ynew, k7);  // FSAL stage: error + interpolation

        // ---- error norm / controller ----
        float ss = 0.0f;
#pragma unroll
        for (int d = 0; d < NDIM; ++d) {
            float err = hc * (C_E1 * k1[d] + C_E2 * k2[d] + C_E3 * k3[d] +
                              C_E4 * k4[d] + C_E5 * k5[d] + C_E6 * k6[d] +
                              C_E7 * k7[d]);
            float scale = C_ATOL + C_RTOL * fmaxf(fabsf(y[d]), fabsf(ynew[d]));
            float q = err / scale;
            ss += q * q;
        }
        const float enorm = sqrtf(ss * (1.0f / NDIM));
        const float enc = fmaxf(enorm, 1e-10f);
        const bool accept = enorm <= 1.0f;
        // enc^-0.2 = exp2(-0.2 * log2(enc)); enc >= 1e-10 (no denorms)
        const float powv =
            __builtin_amdgcn_exp2f(-0.2f * __builtin_amdgcn_logf(enc));
        const float factor = fminf(fmaxf(C_SAFETY * powv, C_FMIN), C_FMAX);
        const bool step_ok = accept && !done;
        const float t_next = step_ok ? tcur + hc : tcur;

        // ---- cubic Hermite interpolation at save points in (t, t_next] ----
        if (step_ok && lane < 16) {
            const float inv = 1.0f / fmaxf(hc, 1e-12f);
            int slo = (int)(tcur * 63.0f) - 1;
            if (slo < 0) slo = 0;
            int shi = (int)((t_next + C_EPS) * 63.0f) + 1;
            if (shi > NSAVE - 1) shi = NSAVE - 1;
            for (int s = slo; s <= shi; ++s) {
                const float te = lds[OFF_TE + s];
                if (te > tcur && te <= t_next + C_EPS) {
                    const float sc = (te - tcur) * inv;
                    const float s2 = sc * sc, s3 = s2 * sc;
                    const float h00 = 2.0f * s3 - 3.0f * s2 + 1.0f;
                    const float h10 = s3 - 2.0f * s2 + sc;
                    const float h01 = -2.0f * s3 + 3.0f * s2;
                    const float h11 = s3 - s2;
#pragma unroll
                    for (int d = 0; d < NDIM; ++d) {
                        float yi = h00 * y[d] + h10 * hc * k1[d] +
                                   h01 * ynew[d] + h11 * hc * k7[d];
                        out[((size_t)bidx * NSAVE + s) * NDIM + d] = yi;
                    }
                }
            }
        }

#pragma unroll
        for (int d = 0; d < NDIM; ++d) y[d] = step_ok ? ynew[d] : y[d];
        tcur = t_next;
        h = done ? h : hc * factor;
    }
}

extern "C" void kernel_launch(void* const* d_in, const int* in_sizes, int n_in,
                              void* d_out, int out_size, void* d_ws,
                              size_t ws_size, hipStream_t stream) {
    const float* y0 = (const float*)d_in[0];
    const float* te = (const float*)d_in[1];
    const float* W1 = (const float*)d_in[2];
    const float* b1 = (const float*)d_in[3];
    const float* W2 = (const float*)d_in[4];
    const float* b2 = (const float*)d_in[5];
    const float* W3 = (const float*)d_in[6];
    const float* b3 = (const float*)d_in[7];
    const float* W4 = (const float*)d_in[8];
    const float* b4 = (const float*)d_in[9];
    float* out = (float*)d_out;

    // 4096 batch / (4 waves * 16 per wave) = 64 workgroups of 128 threads
    NeuralODE_tsit5_wmma_kernel<<<64, 128, LDS_BYTES, stream>>>(
        y0, te, W1, b1, W2, b2, W3, b3, W4, b4, out);
}